// batch_effect_module_5772436046293
// MI455X (gfx1250) — compile-verified
//
#include <hip/hip_runtime.h>

// ---------------------------------------------------------------------------
// out[i, :] = weight[b[i], :]  (N=1M rows, Y_DIM=32 f32), with out[0,:] = 0.
// Bandwidth-bound streaming gather (~132 MB traffic, ~5.7 us floor @ 23.3 TB/s):
//   - TDM (tensor_load_to_lds) stages the 64x32 f32 weight table (8 KB) into
//     LDS once per block; s_wait_tensorcnt + barrier make it visible.
//   - Each thread emits one float4 (16B) => fully coalesced, nontemporal
//     global_store_b128 for the 128 MB write-once output.
//   - 8 lanes share each b[row] load (same cacheline broadcast).
// ---------------------------------------------------------------------------

typedef __attribute__((ext_vector_type(4))) float        f4;
typedef __attribute__((ext_vector_type(4))) unsigned int v4u;
typedef __attribute__((ext_vector_type(8))) int          v8i;
typedef __attribute__((ext_vector_type(4))) int          v4i;

#define NB 64   // N_BATCHES
#define YD 32   // Y_DIM
#define W_F4 (NB * YD / 4)   // 512 float4 = 8 KB

__global__ void batch_effect_gather_kernel(const int* __restrict__ b,
                                           const float* __restrict__ weight,
                                           f4* __restrict__ out4,
                                           int total4) {
    __shared__ f4 lds_w[W_F4];   // sole LDS allocation -> LDS byte offset 0

#if __has_builtin(__builtin_amdgcn_tensor_load_to_lds) && \
    __has_builtin(__builtin_amdgcn_s_wait_tensorcnt)
    // --- TDM path: one wave DMAs the whole 8 KB weight tile into LDS ------
    if (threadIdx.x < 32u) {
        unsigned long long ga = (unsigned long long)(const void*)weight;

        // D# group 0 (4 dwords): count=1, lds_addr=0, global_addr, type=2
        v4u g0 = {
            1u,                                               // count=1, user mode
            0u,                                               // lds_addr = 0
            (unsigned)(ga & 0xFFFFFFFFull),                   // global_addr[31:0]
            ((unsigned)((ga >> 32) & 0x01FFFFFFull)) | (2u << 30) // [56:32] | type=2
        };

        // D# group 1 (8 dwords):
        //  d0: workgroup_mask=0, data_size=2 (4B), no barrier/iterate/pad
        //  tensor_dim0=32 (bits 79:48), tensor_dim1=64 (bits 111:80)
        //  tile_dim0=32 (bits 127:112), tile_dim1=64 (bits 143:128), tile_dim2=0
        //  tensor_dim0_stride=32 (bits 207:160), tensor_dim1_stride=2048 (255:208)
        v8i g1 = {
            (int)(2u << 16),          // data_size = 4 bytes
            (int)((unsigned)YD << 16),// tensor_dim0[15:0] @ bits 63:48
            (int)((unsigned)NB << 16),// tensor_dim0 hi=0; tensor_dim1[15:0] @ 95:80
            (int)((unsigned)YD << 16),// tensor_dim1 hi=0; tile_dim0 @ 127:112
            (int)NB,                  // tile_dim1 @ 143:128; tile_dim2=0
            (int)YD,                  // tensor_dim0_stride[31:0] = 32
            (int)((unsigned)(NB * YD & 0xFFFF) << 16), // dim1_stride[15:0] @ 223:208
            0                         // dim1_stride hi = 0
        };

        v4i gz4 = {0, 0, 0, 0};              // 2D tensor: groups 2/3 unused
        v8i gz8 = {0, 0, 0, 0, 0, 0, 0, 0};  // extra descriptor words (unused)

        // amdgpu-toolchain (clang-23) 6-arg form:
        // (v4u g0, v8i g1, v4i g2, v4i g3, v8i extra, i32 cpol)
        __builtin_amdgcn_tensor_load_to_lds(g0, g1, gz4, gz4, gz8, 0);
        __builtin_amdgcn_s_wait_tensorcnt(0);
    }
    // The TDM builtin carries no pointer operand, so LLVM cannot see that it
    // writes lds_w and would fold the LDS gather to undef (observed: ds=0).
    // Escape the LDS base address + memory clobber => compiler must assume
    // lds_w was written; the gather below stays a real ds_load_b128.
    asm volatile("" : : "s"(&lds_w[0]) : "memory");
#else
    // --- Fallback: cooperative staging copy --------------------------------
    {
        const f4* w4 = (const f4*)weight;
        for (int i = (int)threadIdx.x; i < W_F4; i += (int)blockDim.x)
            lds_w[i] = w4[i];
    }
#endif
    __syncthreads();

    const int gid     = (int)(blockIdx.x * blockDim.x + threadIdx.x);
    const int gstride = (int)(gridDim.x * blockDim.x);

    for (int i = gid; i < total4; i += gstride) {
        const int row  = i >> 3;        // 8 float4 per 32-float row
        const int col4 = i & 7;
        const int batch = b[row];       // 8 lanes broadcast the same value
        f4 v = lds_w[(batch << 3) + col4];
        if (row == 0) v = (f4){0.0f, 0.0f, 0.0f, 0.0f};   // zero first sample
        __builtin_nontemporal_store(v, &out4[i]);         // streaming 16B store
    }
}

extern "C" void kernel_launch(void* const* d_in, const int* in_sizes, int n_in,
                              void* d_out, int out_size, void* d_ws, size_t ws_size,
                              hipStream_t stream) {
    (void)n_in; (void)d_ws; (void)ws_size;
    const int*   b      = (const int*)d_in[0];     // [N,1] int32 batch ids
    const float* weight = (const float*)d_in[1];   // [64,32] f32
    f4*          out4   = (f4*)d_out;              // [N,32] f32 viewed as float4

    const int total4  = out_size / 4;              // number of float4 elements
    const int threads = 256;                       // 8 wave32 per block
    int blocks = (total4 + threads * 8 - 1) / (threads * 8);  // ~8 iters/thread
    if (blocks > 4096) blocks = 4096;
    if (blocks < 1)    blocks = 1;
    (void)in_sizes;

    batch_effect_gather_kernel<<<blocks, threads, 0, stream>>>(b, weight, out4, total4);
}